// TranseUnit_42855183679605
// MI455X (gfx1250) — compile-verified
//
#include <hip/hip_runtime.h>
#include <stdint.h>

// TransE L1 scoring: dist[b,e] = sum_d |(E[s_b]+R[p_b])[d] - E[e,d]|
// BATCH=32, DIM=128, N_ENT=100000, out = 32 x N_ENT f32.
//
// Roofline: ~64 MB HBM traffic (~2.8us floor @ 23.3 TB/s), ~819M f32 VALU ops.
// Abs-inside-reduction -> no matmul structure -> WMMA inapplicable; the CDNA5
// feature that pays here is async global->LDS staging (ASYNCcnt path).

#define DIM     128
#define BATCH   32
#define EPB     256   // entities per block (one per thread)
#define TPB     256   // 8 wave32s
#define ROWPAD  4     // 132-dword LDS row stride: keeps 16B alignment for B128,
                      // cuts 32-way bank conflicts on row reads to 2-way.

#if defined(__has_builtin)
#  if __has_builtin(__builtin_amdgcn_global_load_async_to_lds_b128)
#    define HAVE_ASYNC_BUILTIN 1
#  endif
#  if __has_builtin(__builtin_amdgcn_s_wait_asynccnt)
#    define HAVE_WAIT_BUILTIN 1
#  endif
#endif

typedef int v4i __attribute__((vector_size(4 * sizeof(int))));
typedef __attribute__((address_space(1))) v4i* g_v4i_p;  // global
typedef __attribute__((address_space(3))) v4i* l_v4i_p;  // LDS
typedef __attribute__((address_space(3))) void lds_void;

__device__ __forceinline__ void async_ld_b128(void* lds_dst, const void* gsrc) {
#ifdef HAVE_ASYNC_BUILTIN
  __builtin_amdgcn_global_load_async_to_lds_b128(
      (g_v4i_p)gsrc, (l_v4i_p)lds_dst, /*offset=*/0, /*cpol=*/0);
#else
  unsigned loff = (unsigned)(uintptr_t)(lds_void*)lds_dst;   // LDS byte offset
  unsigned long long g = (unsigned long long)(uintptr_t)gsrc;
  asm volatile("global_load_async_to_lds_b128 %0, %1, off"
               :: "v"(loff), "v"(g) : "memory");
#endif
}

__device__ __forceinline__ void wait_async0() {
#ifdef HAVE_WAIT_BUILTIN
  __builtin_amdgcn_s_wait_asynccnt(0);
#else
  asm volatile("s_wait_asynccnt 0x0" ::: "memory");
#endif
}

__global__ __launch_bounds__(TPB) void transe_l1_kernel(
    const int*   __restrict__ x,     // (32,2) int32 indices
    const float* __restrict__ ent,   // (n_ent,128)
    const float* __restrict__ rel,   // (1000,128)
    float*       __restrict__ out,   // (32,n_ent)
    int n_ent) {
  extern __shared__ float smem[];
  float* qs = smem;                      // 32*128 floats = 16 KB
  float* es = smem + BATCH * DIM;        // 256 rows * 132 floats = 132 KB

  const int t  = threadIdx.x;
  const int e0 = blockIdx.x * EPB;

  // 1) Async-stage this block's entity tile into LDS (coalesced B128s).
  //    Tile = 256 rows * 32 float4; thread t owns float4 indices k*256+t.
  #pragma unroll
  for (int k = 0; k < 32; ++k) {
    int j = k * TPB + t;                 // float4 index in tile [0, 8192)
    int r = j >> 5;                      // tile row
    int c = j & 31;                      // float4 column within row
    int g = min(e0 + r, n_ent - 1);      // clamp tail -> no OOB reads
    const float* gp = ent + (long)g * DIM + c * 4;
    float*       lp = es + r * (DIM + ROWPAD) + c * 4;
    async_ld_b128(lp, gp);
  }

  // 2) Build q = E[s_b] + R[p_b] into LDS; overlaps the async copy.
  for (int i = t; i < BATCH * DIM; i += TPB) {
    int b = i >> 7;
    int d = i & (DIM - 1);
    int s = x[b * 2 + 0];
    int p = x[b * 2 + 1];
    qs[i] = ent[(long)s * DIM + d] + rel[(long)p * DIM + d];
  }

  wait_async0();        // this wave's async loads landed in LDS
  __syncthreads();      // everyone's tile slice + q visible

  // 3) One entity per thread: 32 register accumulators, q via LDS
  //    broadcast (conflict-free), entity row via padded LDS rows.
  const int e = e0 + t;
  if (e < n_ent) {
    const float4* q4 = (const float4*)qs;
    const float*  er = es + t * (DIM + ROWPAD);
    float acc[BATCH];
    #pragma unroll
    for (int b = 0; b < BATCH; ++b) acc[b] = 0.0f;

    #pragma unroll 2
    for (int d4 = 0; d4 < DIM / 4; ++d4) {
      float4 ev = *(const float4*)(er + d4 * 4);
      #pragma unroll
      for (int b = 0; b < BATCH; ++b) {
        float4 qv = q4[b * (DIM / 4) + d4];  // broadcast ds_load_b128
        acc[b] += fabsf(qv.x - ev.x) + fabsf(qv.y - ev.y)
                + fabsf(qv.z - ev.z) + fabsf(qv.w - ev.w);
      }
    }

    // Coalesced stores: consecutive threads -> consecutive e per batch row.
    #pragma unroll
    for (int b = 0; b < BATCH; ++b)
      out[(long)b * n_ent + e] = acc[b];
  }
}

extern "C" void kernel_launch(void* const* d_in, const int* in_sizes, int n_in,
                              void* d_out, int out_size, void* d_ws, size_t ws_size,
                              hipStream_t stream) {
  const int*   x   = (const int*)d_in[0];
  const float* ent = (const float*)d_in[1];
  const float* rel = (const float*)d_in[2];
  float*       out = (float*)d_out;

  const int n_ent = in_sizes[1] / DIM;            // 100000
  const int grid  = (n_ent + EPB - 1) / EPB;      // 391 blocks
  const size_t smem_bytes =
      (size_t)(BATCH * DIM + EPB * (DIM + ROWPAD)) * sizeof(float); // 151552 B

  transe_l1_kernel<<<grid, TPB, smem_bytes, stream>>>(x, ent, rel, out, n_ent);
}